// PretrainingGIN_12000138625368
// MI455X (gfx1250) — compile-verified
//
#include <hip/hip_runtime.h>
#include <hip/hip_bf16.h>

typedef float v2f __attribute__((ext_vector_type(2)));
typedef float v8f __attribute__((ext_vector_type(8)));

#define GIN_D 128
#define LDSS  132   // padded LDS row stride (floats) to avoid bank conflicts

// ---------------------------------------------------------------- copy h->agg
__global__ __launch_bounds__(256) void gin_copy_kernel(const float* __restrict__ in,
                                                       float* __restrict__ outp, int nf4) {
    int i = blockIdx.x * 256 + threadIdx.x;
    if (i < nf4) ((float4*)outp)[i] = ((const float4*)in)[i];
}

// ------------------------------------------------- scatter-add h[src] -> agg[dst]
__global__ __launch_bounds__(256) void gin_scatter_kernel(const float* __restrict__ h,
                                                          const int* __restrict__ src,
                                                          const int* __restrict__ dst,
                                                          float* __restrict__ agg, int ne) {
    int gid  = blockIdx.x * 256 + threadIdx.x;
    int e    = gid >> 5;     // one wave32 per edge
    int lane = gid & 31;     // each lane owns 4 consecutive features
    if (e >= ne) return;
    int s = src[e];
    int d = dst[e];
    float4 v = ((const float4*)(h + (size_t)s * GIN_D))[lane];
    float* p = agg + (size_t)d * GIN_D + lane * 4;
    __hip_atomic_fetch_add(p + 0, v.x, __ATOMIC_RELAXED, __HIP_MEMORY_SCOPE_AGENT);
    __hip_atomic_fetch_add(p + 1, v.y, __ATOMIC_RELAXED, __HIP_MEMORY_SCOPE_AGENT);
    __hip_atomic_fetch_add(p + 2, v.z, __ATOMIC_RELAXED, __HIP_MEMORY_SCOPE_AGENT);
    __hip_atomic_fetch_add(p + 3, v.w, __ATOMIC_RELAXED, __HIP_MEMORY_SCOPE_AGENT);
}

// ------------------- fused MLP: out = [relu](relu(m @ W1 + b1) @ W2 + b2)
// Block = 256 threads = 8 wave32s, handles 16 rows. Wave w computes N-tile w.
// f32 WMMA 16x16x4: A lane layout: lane=(khi*16+m), vgpr j -> A[m][2*khi+j]
//                   B lane layout: lane=(khi*16+n), vgpr j -> B[2*khi+j][n]
//                   C/D: vgpr r, lane l -> [M = r + 8*(l/16)][N = l%16]
__global__ __launch_bounds__(256) void gin_mlp_kernel(const float* __restrict__ m,
                                                      const float* __restrict__ W1p,
                                                      const float* __restrict__ b1p,
                                                      const float* __restrict__ W2p,
                                                      const float* __restrict__ b2p,
                                                      float* __restrict__ outp,
                                                      int n, int relu_out) {
    __shared__ float lds_m[16 * LDSS];
    __shared__ float lds_t[16 * LDSS];
    const int tid  = threadIdx.x;
    const int wave = tid >> 5;
    const int lane = tid & 31;
    const int row0 = blockIdx.x * 16;

    // ---- stage 0: 16x128 input slab -> LDS (256 threads x 2 float4)
#pragma unroll
    for (int it = 0; it < 2; ++it) {
        int idx = tid + it * 256;        // float4 index 0..511
        int r   = idx >> 5;              // 32 float4 per row
        int c4  = idx & 31;
        int gr  = row0 + r;
        if (gr > n - 1) gr = n - 1;      // clamp (keeps EXEC full for WMMA)
        float4 v = ((const float4*)(m + (size_t)gr * GIN_D))[c4];
        float* dl = &lds_m[r * LDSS + c4 * 4];
        dl[0] = v.x; dl[1] = v.y; dl[2] = v.z; dl[3] = v.w;
    }
    __syncthreads();

    const int col   = lane & 15;          // M for A-frag, N for B/C-frag
    const int khalf = (lane >> 4) * 2;    // K sub-offset: 0 or 2
    const int n0    = wave * 16;          // this wave's N-tile base

    // ---- stage 1: T = relu(M @ W1 + b1)
    v8f acc = {};
#pragma unroll 8
    for (int k = 0; k < GIN_D; k += 4) {
        v2f a, b;
        a.x = lds_m[col * LDSS + k + khalf + 0];
        a.y = lds_m[col * LDSS + k + khalf + 1];
        b.x = W1p[(k + khalf + 0) * GIN_D + n0 + col];
        b.y = W1p[(k + khalf + 1) * GIN_D + n0 + col];
        acc = __builtin_amdgcn_wmma_f32_16x16x4_f32(false, a, false, b, (short)0,
                                                    acc, false, false);
    }
    float bias1 = b1p[n0 + col];
#pragma unroll
    for (int r = 0; r < 8; ++r) {
        int rr  = r + ((lane >> 4) << 3);
        float v = acc[r] + bias1;
        lds_t[rr * LDSS + n0 + col] = v > 0.0f ? v : 0.0f;
    }
    __syncthreads();

    // ---- stage 2: O = T @ W2 + b2 (+ optional relu)
    v8f acc2 = {};
#pragma unroll 8
    for (int k = 0; k < GIN_D; k += 4) {
        v2f a, b;
        a.x = lds_t[col * LDSS + k + khalf + 0];
        a.y = lds_t[col * LDSS + k + khalf + 1];
        b.x = W2p[(k + khalf + 0) * GIN_D + n0 + col];
        b.y = W2p[(k + khalf + 1) * GIN_D + n0 + col];
        acc2 = __builtin_amdgcn_wmma_f32_16x16x4_f32(false, a, false, b, (short)0,
                                                     acc2, false, false);
    }
    float bias2 = b2p[n0 + col];
#pragma unroll
    for (int r = 0; r < 8; ++r) {
        int rr = r + ((lane >> 4) << 3);
        int gr = row0 + rr;
        if (gr < n) {
            float v = acc2[r] + bias2;
            if (relu_out) v = v > 0.0f ? v : 0.0f;
            outp[(size_t)gr * GIN_D + n0 + col] = v;
        }
    }
}

// ---------------------------------------------------------------- pooling
__global__ __launch_bounds__(256) void gin_init_out_kernel(float* o, int ntot) {
    int i = blockIdx.x * 256 + threadIdx.x;
    if (i < ntot) o[i] = -__builtin_inff();
}

__device__ inline void atomic_max_float(float* addr, float val) {
    if (val == 0.0f) val = 0.0f;  // canonicalize -0 -> +0
    if (val >= 0.0f) atomicMax((int*)addr, __float_as_int(val));
    else             atomicMin((unsigned int*)addr, __float_as_uint(val));
}

// 128 feature-threads per block; each block scans 64 sorted nodes,
// keeping a register running-max per graph run and flushing atomically.
__global__ __launch_bounds__(128) void gin_pool_kernel(const float* __restrict__ h,
                                                       const int* __restrict__ batch,
                                                       float* __restrict__ o, int n) {
    const int d = threadIdx.x;
    int i0 = blockIdx.x * 64;
    if (i0 >= n) return;
    int iend = i0 + 64; if (iend > n) iend = n;
    int g = batch[i0];
    float mv = -__builtin_inff();
    for (int i = i0; i < iend; ++i) {
        int gi = batch[i];
        if (gi != g) {
            atomic_max_float(&o[g * GIN_D + d], mv);
            g = gi;
            mv = -__builtin_inff();
        }
        float v = h[(size_t)i * GIN_D + d];
        mv = v > mv ? v : mv;
    }
    atomic_max_float(&o[g * GIN_D + d], mv);
}

// ---------------------------------------------------------------- driver
extern "C" void kernel_launch(void* const* d_in, const int* in_sizes, int n_in,
                              void* d_out, int out_size, void* d_ws, size_t ws_size,
                              hipStream_t stream) {
    const float* x     = (const float*)d_in[0];
    const int*   ei    = (const int*)  d_in[1];
    const int*   batch = (const int*)  d_in[2];
    const float* W1    = (const float*)d_in[3];
    const float* b1    = (const float*)d_in[4];
    const float* W2    = (const float*)d_in[5];
    const float* b2    = (const float*)d_in[6];
    float* out = (float*)d_out;

    const int n  = in_sizes[0] / GIN_D;   // 100000
    const int ne = in_sizes[1] / 2;       // 1600000
    const int L  = in_sizes[3] / (GIN_D * GIN_D);  // 3
    const int* src = ei;
    const int* dst = ei + ne;

    float* agg  = (float*)d_ws;                 // [n,128]
    float* hbuf = agg + (size_t)n * GIN_D;      // [n,128]

    const int nf4        = n * (GIN_D / 4);
    const int copyBlocks = (nf4 + 255) / 256;
    const int scatThr    = ne * 32;
    const int scatBlocks = (scatThr + 255) / 256;
    const int mlpBlocks  = (n + 15) / 16;

    const float* h = x;
    for (int l = 0; l < L; ++l) {
        gin_copy_kernel<<<copyBlocks, 256, 0, stream>>>(h, agg, nf4);
        gin_scatter_kernel<<<scatBlocks, 256, 0, stream>>>(h, src, dst, agg, ne);
        gin_mlp_kernel<<<mlpBlocks, 256, 0, stream>>>(
            agg,
            W1 + (size_t)l * GIN_D * GIN_D, b1 + (size_t)l * GIN_D,
            W2 + (size_t)l * GIN_D * GIN_D, b2 + (size_t)l * GIN_D,
            hbuf, n, (l < L - 1) ? 1 : 0);
        h = hbuf;
    }

    gin_init_out_kernel<<<(out_size + 255) / 256, 256, 0, stream>>>(out, out_size);
    gin_pool_kernel<<<(n + 63) / 64, 128, 0, stream>>>(h, batch, out, n);
}